// DinoSwavFusionModel_2267742732812
// MI455X (gfx1250) — compile-verified
//
#include <hip/hip_runtime.h>
#include <hip/hip_bf16.h>
#include <math.h>

#define IMGSZ  224
#define GRID14 14
#define NPATCH 196
#define NTOK   197
#define HEADS  6
#define BATCH  128

typedef __attribute__((ext_vector_type(2))) float v2f;
typedef __attribute__((ext_vector_type(8))) float v8f;

// ---------------------------------------------------------------------------
// Kernel A: per-image attention bbox.
// One 256-thread block (8 wave32) per image.
//   cls (16x16, zero-padded) = mean over heads of attn[b,:,0,-196:]
//   R   (224x16, zero-padded) = fixed bilinear upsample matrix 14 -> 224
//   T   = R * cls            (WMMA f32 16x16x4)
//   amap tile(ty,tx) = T_tile * R_tileT   (WMMA f32 16x16x4)
// Pass 1: max reduction straight from WMMA D fragments.
// Pass 2: recompute tiles, threshold, OR row/col bitmasks -> bbox.
// ---------------------------------------------------------------------------
__global__ __launch_bounds__(256) void dino_bbox_kernel(
    const float* __restrict__ attn, int* __restrict__ box_ws,
    float* __restrict__ bbox_out) {
  __shared__ float cls[16 * 16];
  __shared__ float Rw[IMGSZ * 16];
  __shared__ float Tm[IMGSZ * 16];
  __shared__ float wmax[8];
  __shared__ unsigned rowbits[7];
  __shared__ unsigned colbits[7];
  __shared__ float maxv_sh;
  __shared__ float thr_sh;

  const int b    = blockIdx.x;
  const int tid  = threadIdx.x;
  const int lane = tid & 31;
  const int wave = tid >> 5;

  // ---- init LDS ----
  if (tid < 7) { rowbits[tid] = 0u; colbits[tid] = 0u; }
  cls[tid] = 0.0f;  // 256 entries, blockDim == 256
  __syncthreads();

  // ---- head-mean of CLS attention row (only the 196 patch tokens) ----
  if (tid < NPATCH) {
    float s = 0.0f;
#pragma unroll
    for (int h = 0; h < HEADS; ++h) {
      // attn[b, h, 0, 1 + tid]  (last 196 of 197 columns of row 0)
      s += attn[(size_t)(b * HEADS + h) * NTOK * NTOK + 1 + tid];
    }
    cls[(tid / GRID14) * 16 + (tid % GRID14)] = s * (1.0f / HEADS);
  }

  // ---- build bilinear matrix R (224 x 14, padded to 16 cols) ----
  if (tid < IMGSZ) {
    float s = fmaxf((tid + 0.5f) * (1.0f / 16.0f) - 0.5f, 0.0f);
    int i0 = (int)s;            if (i0 > GRID14 - 1) i0 = GRID14 - 1;
    int i1 = i0 + 1;            if (i1 > GRID14 - 1) i1 = GRID14 - 1;
    float w = s - (float)i0;
#pragma unroll
    for (int k = 0; k < 16; ++k) Rw[tid * 16 + k] = 0.0f;
    Rw[tid * 16 + i0] += 1.0f - w;
    Rw[tid * 16 + i1] += w;
  }
  __syncthreads();

  const int m  = lane & 15;           // row within tile (A) / col (B,D)
  const int kh = (lane >> 4) << 1;    // K half-select for f32 A/B fragments
  const int mh = (lane >> 4) << 3;    // M offset for D fragment rows

  // ---- T = R * cls : 14 row-tiles over 8 waves ----
  for (int t = wave; t < 14; t += 8) {
    v8f acc = {};
#pragma unroll
    for (int kk = 0; kk < 4; ++kk) {
      int k0 = kk * 4 + kh;
      v2f a, bf;
      a.x  = Rw[(t * 16 + m) * 16 + k0];
      a.y  = Rw[(t * 16 + m) * 16 + k0 + 1];
      bf.x = cls[k0 * 16 + m];          // B[k][n], n == lane&15
      bf.y = cls[(k0 + 1) * 16 + m];
      acc = __builtin_amdgcn_wmma_f32_16x16x4_f32(false, a, false, bf,
                                                  (short)0, acc, false, false);
    }
#pragma unroll
    for (int r = 0; r < 8; ++r)
      Tm[(t * 16 + r + mh) * 16 + m] = acc[r];
  }
  __syncthreads();

  // ---- pass 1: amap tiles via WMMA, track max ----
  float lm = -1e30f;
  for (int q = wave; q < 196; q += 8) {
    int ty = q / 14, tx = q % 14;
    v8f acc = {};
#pragma unroll
    for (int kk = 0; kk < 4; ++kk) {
      int k0 = kk * 4 + kh;
      v2f a, bf;
      a.x  = Tm[(ty * 16 + m) * 16 + k0];
      a.y  = Tm[(ty * 16 + m) * 16 + k0 + 1];
      bf.x = Rw[(tx * 16 + m) * 16 + k0];   // (R^T)[k][n] = R[n][k]
      bf.y = Rw[(tx * 16 + m) * 16 + k0 + 1];
      acc = __builtin_amdgcn_wmma_f32_16x16x4_f32(false, a, false, bf,
                                                  (short)0, acc, false, false);
    }
#pragma unroll
    for (int r = 0; r < 8; ++r) lm = fmaxf(lm, acc[r]);
  }
#pragma unroll
  for (int off = 16; off > 0; off >>= 1)
    lm = fmaxf(lm, __shfl_xor(lm, off, 32));
  if (lane == 0) wmax[wave] = lm;
  __syncthreads();
  if (tid == 0) {
    float mv = wmax[0];
#pragma unroll
    for (int i = 1; i < 8; ++i) mv = fmaxf(mv, wmax[i]);
    maxv_sh = mv;
    thr_sh  = 0.6f * mv;
  }
  __syncthreads();
  const float thr = thr_sh;

  // ---- pass 2: recompute tiles, threshold, OR row/col bitmasks ----
  for (int q = wave; q < 196; q += 8) {
    int ty = q / 14, tx = q % 14;
    v8f acc = {};
#pragma unroll
    for (int kk = 0; kk < 4; ++kk) {
      int k0 = kk * 4 + kh;
      v2f a, bf;
      a.x  = Tm[(ty * 16 + m) * 16 + k0];
      a.y  = Tm[(ty * 16 + m) * 16 + k0 + 1];
      bf.x = Rw[(tx * 16 + m) * 16 + k0];
      bf.y = Rw[(tx * 16 + m) * 16 + k0 + 1];
      acc = __builtin_amdgcn_wmma_f32_16x16x4_f32(false, a, false, bf,
                                                  (short)0, acc, false, false);
    }
#pragma unroll
    for (int r = 0; r < 8; ++r) {
      if (acc[r] > thr) {
        int y  = ty * 16 + r + mh;
        int xg = tx * 16 + m;
        atomicOr(&rowbits[y >> 5],  1u << (y & 31));
        atomicOr(&colbits[xg >> 5], 1u << (xg & 31));
      }
    }
  }
  __syncthreads();

  // ---- bbox extraction (single thread; tiny scan) ----
  if (tid == 0) {
    unsigned anyr = 0u;
#pragma unroll
    for (int i = 0; i < 7; ++i) anyr |= rowbits[i] | colbits[i];
    int x1, y1, x2, y2;
    if (maxv_sh < 1e-6f || anyr == 0u) {
      x1 = 0; y1 = 0; x2 = IMGSZ; y2 = IMGSZ;
    } else {
      int ymin = 0, ymax = IMGSZ - 1, xmin = 0, xmax = IMGSZ - 1;
      for (int i = 0; i < IMGSZ; ++i)
        if ((rowbits[i >> 5] >> (i & 31)) & 1u) { ymin = i; break; }
      for (int i = IMGSZ - 1; i >= 0; --i)
        if ((rowbits[i >> 5] >> (i & 31)) & 1u) { ymax = i; break; }
      for (int i = 0; i < IMGSZ; ++i)
        if ((colbits[i >> 5] >> (i & 31)) & 1u) { xmin = i; break; }
      for (int i = IMGSZ - 1; i >= 0; --i)
        if ((colbits[i >> 5] >> (i & 31)) & 1u) { xmax = i; break; }
      int pad_h = (int)floorf((float)(ymax - ymin) * 0.1f);
      int pad_w = (int)floorf((float)(xmax - xmin) * 0.1f);
      x1 = max(0, xmin - pad_w);
      y1 = max(0, ymin - pad_h);
      x2 = min(IMGSZ, xmax + pad_w);
      y2 = min(IMGSZ, ymax + pad_h);
    }
    x2 = max(x2, x1 + 1);
    y2 = max(y2, y1 + 1);
    box_ws[b * 4 + 0] = x1;
    box_ws[b * 4 + 1] = y1;
    box_ws[b * 4 + 2] = x2;
    box_ws[b * 4 + 3] = y2;
    bbox_out[b * 4 + 0] = (float)x1;
    bbox_out[b * 4 + 1] = (float)y1;
    bbox_out[b * 4 + 2] = (float)x2;
    bbox_out[b * 4 + 3] = (float)y2;
  }
}

// ---------------------------------------------------------------------------
// Kernel B: crop + bilinear resize to 224x224. One 224-thread block per
// output row of one (b, c) plane: stores fully coalesced, gathers walk two
// contiguous input-row segments. This stage is the HBM-bound cost center
// (~77 MB read + 77 MB write ~= 6.6 us floor at 23.3 TB/s).
// ---------------------------------------------------------------------------
__global__ __launch_bounds__(IMGSZ) void dino_crop_kernel(
    const float* __restrict__ x, const int* __restrict__ box_ws,
    float* __restrict__ out) {
  const int b  = blockIdx.y;
  const int c  = blockIdx.x / IMGSZ;
  const int y  = blockIdx.x % IMGSZ;
  const int xo = threadIdx.x;

  const int bx1 = box_ws[b * 4 + 0];
  const int by1 = box_ws[b * 4 + 1];
  const int bx2 = box_ws[b * 4 + 2];
  const int by2 = box_ws[b * 4 + 3];
  const int ch = by2 - by1;
  const int cw = bx2 - bx1;

  float sy = fmaxf((y + 0.5f) * ((float)ch * (1.0f / IMGSZ)) - 0.5f, 0.0f);
  int iy0 = (int)sy;  if (iy0 > ch - 1) iy0 = ch - 1;
  int iy1 = iy0 + 1;  if (iy1 > ch - 1) iy1 = ch - 1;
  float wy = sy - (float)iy0;

  float sx = fmaxf((xo + 0.5f) * ((float)cw * (1.0f / IMGSZ)) - 0.5f, 0.0f);
  int ix0 = (int)sx;  if (ix0 > cw - 1) ix0 = cw - 1;
  int ix1 = ix0 + 1;  if (ix1 > cw - 1) ix1 = cw - 1;
  float wx = sx - (float)ix0;

  const float* img = x + (size_t)(b * 3 + c) * IMGSZ * IMGSZ;
  const float* r0  = img + (size_t)(by1 + iy0) * IMGSZ + bx1;
  const float* r1  = img + (size_t)(by1 + iy1) * IMGSZ + bx1;
  __builtin_prefetch(r1 + ix0, 0, 0);  // global_prefetch_b8 on second tap row

  float v00 = r0[ix0];
  float v01 = r0[ix1];
  float v10 = r1[ix0];
  float v11 = r1[ix1];

  float top = v00 + wx * (v01 - v00);
  float bot = v10 + wx * (v11 - v10);
  out[((size_t)(b * 3 + c) * IMGSZ + y) * IMGSZ + xo] = top + wy * (bot - top);
}

extern "C" void kernel_launch(void* const* d_in, const int* in_sizes, int n_in,
                              void* d_out, int out_size, void* d_ws, size_t ws_size,
                              hipStream_t stream) {
  const float* x    = (const float*)d_in[0];  // (128, 3, 224, 224)
  const float* attn = (const float*)d_in[1];  // (128, 6, 197, 197)
  float* out = (float*)d_out;
  int* box_ws = (int*)d_ws;                   // 128 * 4 ints of scratch
  float* bbox_out = out + (size_t)BATCH * 3 * IMGSZ * IMGSZ;

  hipLaunchKernelGGL(dino_bbox_kernel, dim3(BATCH), dim3(256), 0, stream,
                     attn, box_ws, bbox_out);
  hipLaunchKernelGGL(dino_crop_kernel, dim3(3 * IMGSZ, BATCH), dim3(IMGSZ), 0,
                     stream, x, box_ws, out);
}